// c_Attention_3410204033098
// MI455X (gfx1250) — compile-verified
//
#include <hip/hip_runtime.h>
#include <math.h>

// Problem constants (from reference)
#define B_     8
#define N_     1024
#define DIM_   512
#define H_     8
#define DH_    64
#define INNER_ 512          // H*DH
#define SCALE_ 0.125f       // 1/sqrt(64)
#define LNEG_  (-23.025850929940457f)   // -log(1e10)
#define TEMP_INV_ 10.0f     // 1/0.1
#define KREG_  0.3f

typedef float v2f __attribute__((ext_vector_type(2)));
typedef float v8f __attribute__((ext_vector_type(8)));

// D = A(16x4) * B(4x16) + C, fp32 WMMA on gfx1250.
// A per lane: row = lane&15, cols (lane>>4)*2 .. +1   (v2f)
// B per lane: col = lane&15, rows (lane>>4)*2 .. +1   (v2f)
// C/D per lane: elem r -> row r + 8*(lane>>4), col lane&15
__device__ __forceinline__ v8f wmma_f32_16x16x4(v2f a, v2f b, v8f c) {
  return __builtin_amdgcn_wmma_f32_16x16x4_f32(false, a, false, b,
                                               (short)0, c, false, false);
}

// Async global->LDS copy (CDNA5, ASYNCcnt-tracked, bypasses VGPRs).
__device__ __forceinline__ void async_copy_b128(void* lds_dst, const void* gsrc) {
  unsigned lds_b = (unsigned)(uintptr_t)lds_dst;   // LDS aperture: addr[31:0]
  asm volatile("global_load_async_to_lds_b128 %0, %1, off"
               :: "v"(lds_b), "v"(gsrc) : "memory");
}
__device__ __forceinline__ void async_wait0() {
  asm volatile("s_wait_asynccnt 0" ::: "memory");
}

// ---------------------------------------------------------------------------
// Y[m,f] = sum_c X[m,c]*W[f,c] + bias[f],  Cdim fixed at 512 so all inner
// addressing folds into immediate offsets (clauses + staggered waits).
// One wave computes a 64x16 strip (4 accumulators): B-operand reused 4x,
// loads grouped ahead of the 4 WMMAs of each K-step.
// mode 0: Y row-major [M, INNER];  mode 1: qkv permuted [b][h][n][d]
// ---------------------------------------------------------------------------
__global__ __launch_bounds__(32) void gemm_bias_kernel(
    const float* __restrict__ X, const float* __restrict__ W,
    const float* __restrict__ bias, float* __restrict__ Y, int mode) {
  constexpr int Cdim = 512;
  const int Ftiles = INNER_ / 16;
  int tile = blockIdx.x;
  int m0 = (tile / Ftiles) * 64;
  int f0 = (tile % Ftiles) * 16;
  int lane = threadIdx.x;
  int lo = lane & 15, hi = lane >> 4, kb = hi * 2;

  const float* x0 = X + (size_t)(m0 + lo) * Cdim + kb;
  const float* x1 = x0 + 16 * Cdim;
  const float* x2 = x0 + 32 * Cdim;
  const float* x3 = x0 + 48 * Cdim;
  const float* wp = W + (size_t)(f0 + lo) * Cdim + kb;

  v8f acc0 = {}, acc1 = {}, acc2 = {}, acc3 = {};

#pragma unroll 4
  for (int c0 = 0; c0 < Cdim; c0 += 4) {
    v2f b  = *(const v2f*)(wp + c0);
    v2f a0 = *(const v2f*)(x0 + c0);
    v2f a1 = *(const v2f*)(x1 + c0);
    v2f a2 = *(const v2f*)(x2 + c0);
    v2f a3 = *(const v2f*)(x3 + c0);
    acc0 = wmma_f32_16x16x4(a0, b, acc0);
    acc1 = wmma_f32_16x16x4(a1, b, acc1);
    acc2 = wmma_f32_16x16x4(a2, b, acc2);
    acc3 = wmma_f32_16x16x4(a3, b, acc3);
  }

  int f = f0 + lo;
  float bvv = bias[f];
  auto store_tile = [&](const v8f& a, int t) {
#pragma unroll
    for (int r = 0; r < 8; ++r) {
      int row = m0 + t * 16 + r + 8 * hi;
      float val = a[r] + bvv;
      if (mode == 0) {
        Y[(size_t)row * INNER_ + f] = val;
      } else {
        int bb = row >> 10, nn = row & (N_ - 1);
        int h = f >> 6, d = f & (DH_ - 1);
        Y[(((size_t)bb * H_ + h) * N_ + nn) * DH_ + d] = val;
      }
    }
  };
  store_tile(acc0, 0);
  store_tile(acc1, 1);
  store_tile(acc2, 2);
  store_tile(acc3, 3);
}

// ---------------------------------------------------------------------------
// Per-(b,h,n) L2-normalize 64-dim head vectors; repack to [b][n][h*64+d]
// so the head-averaged cosine becomes one 512-deep GEMM. One wave per row.
// ---------------------------------------------------------------------------
__global__ __launch_bounds__(128) void normalize_kernel(
    const float* __restrict__ src, float* __restrict__ dst) {
  int row = blockIdx.x * 4 + (threadIdx.x >> 5);   // row in [0, B*H*N)
  int lane = threadIdx.x & 31;
  int bb = row / (H_ * N_);
  int rem = row % (H_ * N_);
  int h = rem / N_, nn = rem % N_;
  const float* p = src + (size_t)row * DH_;
  float x0 = p[lane], x1 = p[lane + 32];
  float ss = x0 * x0 + x1 * x1;
#pragma unroll
  for (int off = 16; off > 0; off >>= 1) ss += __shfl_xor(ss, off);
  float inv = 1.0f / sqrtf(ss);
  size_t base = ((size_t)bb * N_ + nn) * INNER_ + (size_t)h * DH_;
  dst[base + lane] = x0 * inv;
  dst[base + lane + 32] = x1 * inv;
}

// ---------------------------------------------------------------------------
// Fused attention per (b, h, 16-row tile):
//  phase1: scores = Q Kt (WMMA; Q strip in regs, K tile loads batched ahead
//          of the 16 chained WMMAs), mask+scale into LDS (16x1024 f32)
//  phase2: stable softmax via 8-lane shuffle reductions, write attn (float4)
//  phase3: out = attn @ V (WMMA; grouped LDS/global loads) -> AO
// 4 waves/block; each wave owns 256 score cols (phase1), 4 score rows (phase2), 16 out cols (3).
// ---------------------------------------------------------------------------
__global__ __launch_bounds__(128) void attention_kernel(
    const float* __restrict__ Q, const float* __restrict__ K,
    const float* __restrict__ V, const float* __restrict__ mask,
    float* __restrict__ attn_out, float* __restrict__ AO) {
  __shared__ float s[16 * N_];      // 64 KB score strip

  int bh = blockIdx.x >> 6;         // b*H + h
  int i0 = (blockIdx.x & 63) * 16;
  int bb = bh >> 3, h = bh & 7;
  const float* Qm = Q + (size_t)bh * N_ * DH_;
  const float* Km = K + (size_t)bh * N_ * DH_;
  const float* Vm = V + (size_t)bh * N_ * DH_;
  const float* mb = mask + (size_t)bb * N_ * N_;

  int tid = threadIdx.x;
  int wave = tid >> 5, lane = tid & 31;
  int lo = lane & 15, hi = lane >> 4, kb = hi * 2;

  // ---- phase 1: masked, scaled scores into LDS ----
  v2f qreg[16];                      // this wave's Q strip, loaded once
  {
    const float* qrow = Qm + (size_t)(i0 + lo) * DH_ + kb;
#pragma unroll
    for (int t = 0; t < 16; ++t) qreg[t] = *(const v2f*)(qrow + 4 * t);
  }
  for (int jt = 0; jt < 16; ++jt) {
    int j0 = wave * 256 + jt * 16;
    const float* krow = Km + (size_t)(j0 + lo) * DH_ + kb;
    __builtin_prefetch(krow + 16 * DH_, 0, 1);   // next j-tile's K rows
    v2f breg[16];
#pragma unroll
    for (int t = 0; t < 16; ++t) breg[t] = *(const v2f*)(krow + 4 * t);
    v8f acc = {};
#pragma unroll
    for (int t = 0; t < 16; ++t) acc = wmma_f32_16x16x4(qreg[t], breg[t], acc);
#pragma unroll
    for (int r = 0; r < 8; ++r) {
      int rr = r + 8 * hi;
      int gi = i0 + rr, col = j0 + lo;
      float mv = mb[(size_t)gi * N_ + col];
      float vsc = acc[r] * mv + (mv == 0.0f ? LNEG_ : 0.0f);
      s[rr * N_ + col] = vsc * SCALE_;
    }
  }
  __syncthreads();

  // ---- phase 2: softmax per row (8 threads/row), write attn to d_out ----
  {
    int row = tid >> 3, sub = tid & 7;
    float* sr = s + row * N_ + sub * 128;
    float mx = -3.4e38f;
    for (int t = 0; t < 128; ++t) mx = fmaxf(mx, sr[t]);
#pragma unroll
    for (int off = 1; off < 8; off <<= 1) mx = fmaxf(mx, __shfl_xor(mx, off));
    float sum = 0.0f;
    for (int t = 0; t < 128; ++t) {
      float e = expf(sr[t] - mx);
      sr[t] = e;
      sum += e;
    }
#pragma unroll
    for (int off = 1; off < 8; off <<= 1) sum += __shfl_xor(sum, off);
    float inv = 1.0f / sum;
    float* gout = attn_out + ((size_t)bh * N_ + (i0 + row)) * N_ + sub * 128;
    for (int t = 0; t < 128; t += 4) {
      float4 e4 = *(float4*)(sr + t);
      e4.x *= inv; e4.y *= inv; e4.z *= inv; e4.w *= inv;
      *(float4*)(sr + t) = e4;
      *(float4*)(gout + t) = e4;
    }
  }
  __syncthreads();

  // ---- phase 3: out = attn @ V  (A from LDS; loads grouped per 4 WMMAs) ----
  {
    int c0v = wave * 16;
    const float* vcol = Vm + c0v + lo;
    v8f acc = {};
    const float* srow = s + lo * N_ + kb;      // A row = lane&15
    for (int k0 = 0; k0 < N_; k0 += 16) {
      __builtin_prefetch(vcol + (size_t)(k0 + 16 + kb) * DH_, 0, 1);
      v2f a[4], b[4];
#pragma unroll
      for (int u = 0; u < 4; ++u) {
        a[u] = *(const v2f*)(srow + k0 + 4 * u);           // ds_load_b64
        b[u].x = vcol[(size_t)(k0 + 4 * u + kb) * DH_];
        b[u].y = vcol[(size_t)(k0 + 4 * u + kb + 1) * DH_];
      }
#pragma unroll
      for (int u = 0; u < 4; ++u) acc = wmma_f32_16x16x4(a[u], b[u], acc);
    }
#pragma unroll
    for (int r = 0; r < 8; ++r) {
      int row = i0 + r + 8 * hi;
      AO[((size_t)bb * N_ + row) * INNER_ + h * DH_ + c0v + lo] = acc[r];
    }
  }
}

// ---------------------------------------------------------------------------
// cosine[b,i,j] = (1/H) * qhat[b,i,:] . khat[b,j,:]   (512-deep WMMA GEMM)
// clustering_i = -log( sum_j e*mask / sum_j e ),  e = exp(cos/TEMP)
// The 16x512 Qhat tile is staged into LDS ONCE via async global->LDS b128
// copies (ASYNCcnt), then re-read 16x as the WMMA A-operand via ds_load_b64.
// ---------------------------------------------------------------------------
__global__ __launch_bounds__(128) void cosine_loss_kernel(
    const float* __restrict__ QH, const float* __restrict__ KH,
    const float* __restrict__ mask, float* __restrict__ loss) {
  __shared__ float sQ[16 * INNER_];   // 32 KB staged Qhat tile
  __shared__ float redA[128][8];      // 4 KB
  __shared__ float redM[128][8];      // 4 KB

  int bb = blockIdx.x >> 6;
  int i0 = (blockIdx.x & 63) * 16;
  const float* Qh = QH + (size_t)bb * N_ * INNER_;
  const float* Kh = KH + (size_t)bb * N_ * INNER_;
  const float* mb = mask + (size_t)bb * N_ * N_;
  int tid = threadIdx.x;
  int wave = tid >> 5, lane = tid & 31;
  int lo = lane & 15, hi = lane >> 4, kb = hi * 2;

  // cooperative async copy: rows i0..i0+15 are one contiguous 32KB block
  {
    const float* gq = Qh + (size_t)i0 * INNER_;
    for (int off = tid * 4; off < 16 * INNER_; off += 128 * 4)
      async_copy_b128(sQ + off, gq + off);
    async_wait0();
  }
  __syncthreads();

  float accA[8], accM[8];
#pragma unroll
  for (int r = 0; r < 8; ++r) { accA[r] = 0.0f; accM[r] = 0.0f; }

  const float* qlds = sQ + (size_t)lo * INNER_ + kb;   // A-operand from LDS
  for (int jt = 0; jt < 16; ++jt) {
    int j0 = wave * 256 + jt * 16;
    const float* krow = Kh + (size_t)(j0 + lo) * INNER_ + kb;
    __builtin_prefetch(krow + 16 * INNER_, 0, 1);
    v8f acc = {};
    for (int c0 = 0; c0 < INNER_; c0 += 16) {
      v2f a[4], b[4];
#pragma unroll
      for (int u = 0; u < 4; ++u) {
        a[u] = *(const v2f*)(qlds + c0 + 4 * u);          // ds_load_b64
        b[u] = *(const v2f*)(krow + c0 + 4 * u);
      }
#pragma unroll
      for (int u = 0; u < 4; ++u) acc = wmma_f32_16x16x4(a[u], b[u], acc);
    }
#pragma unroll
    for (int r = 0; r < 8; ++r) {
      int gi = i0 + r + 8 * hi, col = j0 + lo;
      float e = expf(acc[r] * (1.0f / H_) * TEMP_INV_);   // non-stabilized, per ref
      accA[r] += e;
      accM[r] += e * mb[(size_t)gi * N_ + col];
    }
  }
#pragma unroll
  for (int r = 0; r < 8; ++r) { redA[tid][r] = accA[r]; redM[tid][r] = accM[r]; }
  __syncthreads();

  if (tid < 16) {                      // one thread per tile row
    int rloc = tid & 7, half = tid >> 3;
    float sA = 0.0f, sM = 0.0f;
    for (int w = 0; w < 4; ++w)
      for (int l = 0; l < 16; ++l) {
        int idx = w * 32 + half * 16 + l;
        sA += redA[idx][rloc];
        sM += redM[idx][rloc];
      }
    float clustering = -logf(sM / sA);
    atomicAdd(loss, clustering * (1.0f / ((float)N_ * (float)B_)));
  }
}

__global__ void init_loss_kernel(float* loss) {
  if (threadIdx.x == 0 && blockIdx.x == 0) *loss = 0.0f;
}

// KREG * mean_b [ (1/(n(n-1))) * sum |eye - mask| ]
__global__ __launch_bounds__(256) void regular_loss_kernel(
    const float* __restrict__ mask, float* __restrict__ loss) {
  __shared__ float red[256];
  const size_t total = (size_t)B_ * N_ * N_;
  float acc = 0.0f;
  for (size_t e = (size_t)blockIdx.x * blockDim.x + threadIdx.x; e < total;
       e += (size_t)gridDim.x * blockDim.x) {
    size_t rem = e % ((size_t)N_ * N_);
    int i = (int)(rem / N_), j = (int)(rem % N_);
    float eye = (i == j) ? 1.0f : 0.0f;
    acc += fabsf(eye - mask[e]);
  }
  red[threadIdx.x] = acc;
  __syncthreads();
  for (int off = 128; off > 0; off >>= 1) {
    if (threadIdx.x < (unsigned)off) red[threadIdx.x] += red[threadIdx.x + off];
    __syncthreads();
  }
  if (threadIdx.x == 0)
    atomicAdd(loss, red[0] * (KREG_ / ((float)B_ * (float)N_ * (float)(N_ - 1))));
}

// ---------------------------------------------------------------------------
extern "C" void kernel_launch(void* const* d_in, const int* in_sizes, int n_in,
                              void* d_out, int out_size, void* d_ws, size_t ws_size,
                              hipStream_t stream) {
  (void)in_sizes; (void)n_in; (void)out_size; (void)ws_size;
  const float* x    = (const float*)d_in[0];
  const float* mask = (const float*)d_in[1];
  const float* wq   = (const float*)d_in[2];
  const float* bq   = (const float*)d_in[3];
  const float* wk   = (const float*)d_in[4];
  const float* bk   = (const float*)d_in[5];
  const float* wv   = (const float*)d_in[6];
  const float* bv   = (const float*)d_in[7];
  const float* wo   = (const float*)d_in[8];
  const float* bo   = (const float*)d_in[9];

  float* out  = (float*)d_out;                          // [B,N,512]
  float* attn = out + (size_t)B_ * N_ * INNER_;         // [B,H,N,N]
  float* loss = attn + (size_t)B_ * H_ * N_ * N_;       // scalar

  const size_t QKV = (size_t)B_ * H_ * N_ * DH_;        // 4,194,304 floats
  float* ws = (float*)d_ws;
  float* Q  = ws;
  float* K  = Q  + QKV;
  float* V  = K  + QKV;
  float* QH = V  + QKV;                                 // [B,N,512] normalized
  float* KH = QH + QKV;
  float* AO = KH + QKV;                                 // [B,N,512] attn output

  const int M = B_ * N_;                                // 8192
  const int gemm_blocks = (M / 64) * (INNER_ / 16);     // 4096 waves, 64x16 each

  // QKV projections (WMMA f32)
  gemm_bias_kernel<<<gemm_blocks, 32, 0, stream>>>(x, wq, bq, Q, 1);
  gemm_bias_kernel<<<gemm_blocks, 32, 0, stream>>>(x, wk, bk, K, 1);
  gemm_bias_kernel<<<gemm_blocks, 32, 0, stream>>>(x, wv, bv, V, 1);

  // head-wise normalization for cosine GEMM
  normalize_kernel<<<(B_ * H_ * N_) / 4, 128, 0, stream>>>(Q, QH);
  normalize_kernel<<<(B_ * H_ * N_) / 4, 128, 0, stream>>>(K, KH);

  // fused scores -> softmax -> attn write -> attn@V
  attention_kernel<<<B_ * H_ * (N_ / 16), 128, 0, stream>>>(Q, K, V, mask, attn, AO);

  // output projection
  gemm_bias_kernel<<<gemm_blocks, 32, 0, stream>>>(AO, wo, bo, out, 0);

  // loss = clustering.mean + KREG * regular (accumulated atomically)
  init_loss_kernel<<<1, 1, 0, stream>>>(loss);
  regular_loss_kernel<<<1024, 256, 0, stream>>>(mask, loss);
  cosine_loss_kernel<<<B_ * (N_ / 16), 128, 0, stream>>>(QH, KH, mask, loss);
}